// _MaxPoolHex_42880953483674
// MI455X (gfx1250) — compile-verified
//
#include <hip/hip_runtime.h>

// MaxPoolHex on gfx1250: stream rows through LDS via async global->LDS copies
// (CDNA5 ASYNCcnt path), compute 7-tap hex max from LDS, masked b64 stores.
//
// x:   (8,32,512,512) fp32  -> 256 images of 512x512
// out: (8,32,256,256) fp32
//
// K=3, S=2, P=1, PAD_VAL=1e-9. Hex taps = 3x3 minus (0,0),(2,2).
// Anti-diagonal (r+c==512) of the unpadded image replaced by PAD_VAL.
// Output zeroed where (oi+oj) >= 256.

#define W           512
#define HOUT        256
#define TILE_ROWS   8                      // output rows per block
#define IN_ROWS     (2 * TILE_ROWS + 1)    // 17 input rows staged per tile
#define LDS_ROW_F   (W + 4)                // pad 4 floats (keeps 16B alignment)
#define PAD_VALF    1e-9f

__global__ __launch_bounds__(256)
void _MaxPoolHex_42880953483674_kernel(const float* __restrict__ x,
                                       float* __restrict__ out) {
    __shared__ float smem[IN_ROWS * LDS_ROW_F];

    const int tid  = threadIdx.x;          // 256 threads = 8 wave32
    const int tile = blockIdx.x;           // img*32 + row_tile
    const int img  = tile >> 5;
    const int ro   = (tile & 31) * TILE_ROWS;   // first output row of tile
    const int gr0  = 2 * ro - 1;                // first unpadded input row (may be -1)
    const int in_base  = img * (W * W);
    const int out_base = img * (HOUT * HOUT);

    // ---- Stage input rows [gr0, gr0+16] into LDS with async B128 copies ----
    // 17 rows * 128 chunks of 16B = 2176 chunks; 256 threads -> ~8.5 rounds.
    const int TOTAL_CHUNKS = IN_ROWS * (W / 4);
    for (int chunk = tid; chunk < TOTAL_CHUNKS; chunk += 256) {
        const int lr  = chunk >> 7;        // local row 0..16
        const int c16 = chunk & 127;       // 16B chunk within row
        const int gr  = gr0 + lr;          // unpadded input row
        const unsigned lds_off =
            (unsigned)((lr * LDS_ROW_F + c16 * 4) * 4);   // byte offset in LDS
        if ((unsigned)gr < (unsigned)W) {
            const float* gptr = x + in_base + gr * W + c16 * 4;
            asm volatile("global_load_async_to_lds_b128 %0, %1, off"
                         :: "v"(lds_off), "v"(gptr)
                         : "memory");
        } else {
            // top/bottom padding rows: fill with PAD_VAL (ds_store, immediate in VGPR)
            float* p = &smem[lr * LDS_ROW_F + c16 * 4];
            p[0] = PAD_VALF; p[1] = PAD_VALF; p[2] = PAD_VALF; p[3] = PAD_VALF;
        }
    }
    asm volatile("s_wait_asynccnt 0x0" ::: "memory");
    __syncthreads();

    // ---- Compute: thread = (column pair cp, row group rg) ----
    // cols j0=2cp, j1=2cp+1 ; output rows ro + rg*4 + (0..3)
    const int cp = tid & 127;
    const int rg = tid >> 7;
    const int r0 = rg * 4;                 // first local output row
    const int c0 = 4 * cp;                 // base unpadded col (0..508)

    // Cache the 9 LDS rows this thread needs: lr = 2*r0 + k, k=0..8.
    // Needed cols: c0-1 (guarded) and c0..c0+3 (always in range -> one b128).
    float am1[9], a0[9], a1[9], a2[9], a3[9];
#pragma unroll
    for (int k = 0; k < 9; ++k) {
        const int lr   = 2 * r0 + k;
        const int urow = gr0 + lr;         // unpadded input row (-1..512)
        const float* rowp = &smem[lr * LDS_ROW_F];
        const float4 q = *(const float4*)&rowp[c0];
        float vm1 = (c0 == 0) ? PAD_VALF : rowp[c0 - 1];
        const int d = W - urow;            // anti-diagonal column for this row
        am1[k] = (c0 - 1 == d) ? PAD_VALF : vm1;
        a0[k]  = (c0     == d) ? PAD_VALF : q.x;
        a1[k]  = (c0 + 1 == d) ? PAD_VALF : q.y;
        a2[k]  = (c0 + 2 == d) ? PAD_VALF : q.z;
        a3[k]  = (c0 + 3 == d) ? PAD_VALF : q.w;
    }

#pragma unroll
    for (int r = 0; r < 4; ++r) {
        const int og = ro + r0 + r;        // global output row
        const int ka = 2 * r, kb = 2 * r + 1, kc = 2 * r + 2;

        // col j0 = 2cp: taps (0,1)(0,2)=a0,a1[ka]; (1,0)(1,1)(1,2)=am1,a0,a1[kb];
        //               (2,0)(2,1)=am1,a0[kc]
        float m0 = fmaxf(a0[ka], a1[ka]);
        m0 = fmaxf(m0, fmaxf(am1[kb], fmaxf(a0[kb], a1[kb])));
        m0 = fmaxf(m0, fmaxf(am1[kc], a0[kc]));

        // col j1 = 2cp+1: taps (0,1)(0,2)=a2,a3[ka]; (1,*)=a1,a2,a3[kb];
        //                 (2,0)(2,1)=a1,a2[kc]
        float m1 = fmaxf(a2[ka], a3[ka]);
        m1 = fmaxf(m1, fmaxf(a1[kb], fmaxf(a2[kb], a3[kb])));
        m1 = fmaxf(m1, fmaxf(a1[kc], a2[kc]));

        // triangle output mask: zero where og + col >= 256
        float2 o;
        o.x = (og + 2 * cp     < HOUT) ? m0 : 0.0f;
        o.y = (og + 2 * cp + 1 < HOUT) ? m1 : 0.0f;
        *(float2*)&out[out_base + og * HOUT + 2 * cp] = o;
    }
}

extern "C" void kernel_launch(void* const* d_in, const int* in_sizes, int n_in,
                              void* d_out, int out_size, void* d_ws, size_t ws_size,
                              hipStream_t stream) {
    const float* x = (const float*)d_in[0];
    float* out     = (float*)d_out;

    const int images = in_sizes[0] / (W * W);            // 8*32 = 256
    const int blocks = images * (HOUT / TILE_ROWS);      // 256*32 = 8192

    _MaxPoolHex_42880953483674_kernel<<<dim3(blocks), dim3(256), 0, stream>>>(x, out);
}